// AttentionLayer_27419071218010
// MI455X (gfx1250) — compile-verified
//
#include <hip/hip_runtime.h>
#include <math.h>

// Problem constants (from reference): B=32, T=8192, D=H=128
#define BB 32
#define TT 8192
#define DD 128
#define LSTRIDE 132   // padded LDS row stride (floats): banks (4n+k)%64, conflict-free

typedef __attribute__((ext_vector_type(2))) float v2f;
typedef __attribute__((ext_vector_type(8))) float v8f;

// ---------------------------------------------------------------------------
// K1: bias[b,d] = sum_h dec[b,h] * W2[h,d]     (32 x 128 outputs, trivial)
// ---------------------------------------------------------------------------
__global__ __launch_bounds__(128) void bias_kernel(const float* __restrict__ dec,
                                                   const float* __restrict__ W2,
                                                   float* __restrict__ bias) {
    int b = blockIdx.x, d = threadIdx.x;
    float acc = 0.f;
#pragma unroll 8
    for (int h = 0; h < DD; ++h)
        acc += dec[b * DD + h] * W2[h * DD + d];
    bias[b * DD + d] = acc;
}

// ---------------------------------------------------------------------------
// K2: logits[b,t] = V . tanh(enc[b,t,:] @ W1 + bias[b,:])
// Fused fp32 WMMA GEMM + tanh + V-dot. One wave per 16-row t-strip.
// W1 staged transposed in LDS with padded stride 132 -> linear addresses:
// inner loop is ds_load_b64(imm offset) + v_wmma_f32_16x16x4_f32 only.
// ---------------------------------------------------------------------------
__global__ __launch_bounds__(256) void score_kernel(const float* __restrict__ enc,
                                                    const float* __restrict__ W1,
                                                    const float* __restrict__ V,
                                                    const float* __restrict__ bias,
                                                    float* __restrict__ logits) {
    __shared__ float w1t[DD * LSTRIDE];  // 67584 bytes: w1t[n][k] transposed

    const int b    = blockIdx.x >> 6;       // 64 row-tiles of 128 per batch
    const int tile = blockIdx.x & 63;
    const int t0   = tile * 128;
    const int tid  = threadIdx.x;

    // Stage W1 transposed into LDS (w1t[n][k] = W1[k][n]).
    for (int i = tid; i < DD * DD; i += 256) {
        int k = i >> 7, n = i & 127;
        w1t[n * LSTRIDE + k] = W1[i];
    }
    __syncthreads();

    const int wave = tid >> 5;
    const int lane = tid & 31;
    const int half = lane >> 4;      // A/B operand K-half select (per ISA layout)
    const int l16  = lane & 15;      // A: row M; B/D: column N

    // Preload A strip: rows t0+wave*16+l16; lanes 0-15 carry K={4k,4k+1},
    // lanes 16-31 carry K={4k+2,4k+3}.
    const float* arow = enc + ((size_t)b * TT + (size_t)(t0 + wave * 16 + l16)) * DD;
    v2f a[32];
#pragma unroll
    for (int kk = 0; kk < 32; ++kk)
        a[kk] = *(const v2f*)(arow + 4 * kk + 2 * half);

    // Hoist bias/V fetches out of the hot loop (per-lane column nt*16+l16).
    float bias8[8], vv8[8];
#pragma unroll
    for (int nt = 0; nt < 8; ++nt) {
        bias8[nt] = bias[b * DD + nt * 16 + l16];
        vv8[nt]   = V[nt * 16 + l16];
    }

    float plog[8];
#pragma unroll
    for (int j = 0; j < 8; ++j) plog[j] = 0.f;

    // 8 N-tiles of 16 columns; K=128 accumulated via 32 wmma_f32_16x16x4 each.
    for (int nt = 0; nt < 8; ++nt) {
        const int ncol = nt * 16 + l16;
        const float* bptr = &w1t[ncol * LSTRIDE + 2 * half];
        v8f c = {};
#pragma unroll
        for (int kk = 0; kk < 32; ++kk) {
            v2f bb = *(const v2f*)(bptr + 4 * kk);   // imm-offset ds_load_b64
            c = __builtin_amdgcn_wmma_f32_16x16x4_f32(
                    false, a[kk], false, bb, (short)0, c, false, false);
        }
        const float bias_n = bias8[nt];
        const float v_n    = vv8[nt];
#pragma unroll
        for (int j = 0; j < 8; ++j)  // lane holds D[M=j+8*half, N=ncol]
            plog[j] += tanhf(c[j] + bias_n) * v_n;
    }

    // Reduce over the 16 lanes of each half (columns) -> full 128-col dot.
#pragma unroll
    for (int off = 1; off < 16; off <<= 1)
#pragma unroll
        for (int j = 0; j < 8; ++j)
            plog[j] += __shfl_xor(plog[j], off, 32);

    if (l16 == 0) {
        float* lp = logits + (size_t)b * TT + t0 + wave * 16 + half * 8;
#pragma unroll
        for (int j = 0; j < 8; ++j) lp[j] = plog[j];
    }
}

// ---------------------------------------------------------------------------
// K3: per-batch softmax stats over T: m[b]=max, s[b]=sum(exp(l-m))
// ---------------------------------------------------------------------------
__global__ __launch_bounds__(256) void softmax_stats(const float* __restrict__ logits,
                                                     float* __restrict__ mbuf,
                                                     float* __restrict__ sbuf) {
    const int b = blockIdx.x, tid = threadIdx.x;
    const int wave = tid >> 5, lane = tid & 31;
    const float* lp = logits + (size_t)b * TT;
    __shared__ float redm[8];
    __shared__ float reds[8];

    float m = -3.402823466e+38f;
    for (int t = tid; t < TT; t += 256) m = fmaxf(m, lp[t]);
#pragma unroll
    for (int off = 16; off; off >>= 1) m = fmaxf(m, __shfl_xor(m, off, 32));
    if (lane == 0) redm[wave] = m;
    __syncthreads();
    m = redm[0];
#pragma unroll
    for (int i = 1; i < 8; ++i) m = fmaxf(m, redm[i]);

    float s = 0.f;
    for (int t = tid; t < TT; t += 256) s += expf(lp[t] - m);
#pragma unroll
    for (int off = 16; off; off >>= 1) s += __shfl_xor(s, off, 32);
    if (lane == 0) reds[wave] = s;
    __syncthreads();
    if (tid == 0) {
        float tot = 0.f;
#pragma unroll
        for (int i = 0; i < 8; ++i) tot += reds[i];
        mbuf[b] = m;
        sbuf[b] = tot;
    }
}

// ---------------------------------------------------------------------------
// K4: weights = exp(l-m)/s -> d_out ; per-chunk partial context -> ws
// One block = one batch x 256 t-rows.
// ---------------------------------------------------------------------------
__global__ __launch_bounds__(256) void finalize_kernel(const float* __restrict__ enc,
                                                       const float* __restrict__ logits,
                                                       const float* __restrict__ mbuf,
                                                       const float* __restrict__ sbuf,
                                                       float* __restrict__ out,
                                                       float* __restrict__ part) {
    const int b     = blockIdx.x >> 5;   // 32 chunks of 256 rows per batch
    const int chunk = blockIdx.x & 31;
    const int t0    = chunk * 256;
    const int tid   = threadIdx.x;
    __shared__ float wsh[256];
    __shared__ float acc2[256];

    const float m = mbuf[b], sinv = 1.0f / sbuf[b];
    const float w = expf(logits[(size_t)b * TT + t0 + tid] - m) * sinv;
    wsh[tid] = w;
    out[BB * DD + (size_t)b * TT + t0 + tid] = w;   // attention_weights
    __syncthreads();

    const int d = tid & 127, g = tid >> 7;          // 2 row-groups x 128 d-lanes
    float acc = 0.f;
    const float* ep = enc + ((size_t)b * TT + t0) * DD + d;
    for (int r = g; r < 256; r += 2)
        acc += wsh[r] * ep[(size_t)r * DD];
    acc2[tid] = acc;
    __syncthreads();
    if (tid < 128)
        part[((size_t)b * 32 + chunk) * DD + tid] = acc2[tid] + acc2[tid + 128];
}

// ---------------------------------------------------------------------------
// K5: context[b,d] = sum over 32 chunk partials (deterministic order)
// ---------------------------------------------------------------------------
__global__ __launch_bounds__(128) void ctx_reduce(const float* __restrict__ part,
                                                  float* __restrict__ out) {
    const int b = blockIdx.x, d = threadIdx.x;
    float acc = 0.f;
#pragma unroll 8
    for (int c = 0; c < 32; ++c)
        acc += part[((size_t)b * 32 + c) * DD + d];
    out[b * DD + d] = acc;   // context_vector at front of d_out
}

// ---------------------------------------------------------------------------
extern "C" void kernel_launch(void* const* d_in, const int* in_sizes, int n_in,
                              void* d_out, int out_size, void* d_ws, size_t ws_size,
                              hipStream_t stream) {
    const float* enc = (const float*)d_in[0];  // [B,T,D]
    const float* dec = (const float*)d_in[1];  // [B,H]
    const float* W1  = (const float*)d_in[2];  // [D,D]
    const float* W2  = (const float*)d_in[3];  // [H,D]
    const float* V   = (const float*)d_in[4];  // [D,1]
    float* out = (float*)d_out;                // context (4096) ++ weights (262144)

    float* ws     = (float*)d_ws;
    float* bias   = ws;                        // B*D          = 4096
    float* logits = bias + BB * DD;            // B*T          = 262144
    float* mbuf   = logits + (size_t)BB * TT;  // B            = 32
    float* sbuf   = mbuf + BB;                 // B            = 32
    float* part   = sbuf + BB;                 // B*32*D       = 131072

    bias_kernel   <<<BB, 128, 0, stream>>>(dec, W2, bias);
    score_kernel  <<<BB * (TT / 128), 256, 0, stream>>>(enc, W1, V, bias, logits);
    softmax_stats <<<BB, 256, 0, stream>>>(logits, mbuf, sbuf);
    finalize_kernel<<<BB * (TT / 256), 256, 0, stream>>>(enc, logits, mbuf, sbuf, out, part);
    ctx_reduce    <<<BB, 128, 0, stream>>>(part, out);
}